// MyGAT_89043261981500
// MI455X (gfx1250) — compile-verified
//
#include <hip/hip_runtime.h>
#include <math.h>

typedef __attribute__((ext_vector_type(2))) float v2f;
typedef __attribute__((ext_vector_type(8))) float v8f;

#define LDA 132  // padded LDS stride: 132 % 64 = 4 -> conflict-free column reads

// ---------------- GEMM: C[M,NC] = A[M,128] @ B[128,NC], fp32 WMMA ----------------
// NC is compile-time so B addressing folds to a per-lane base + immediate offsets.
template <int NC>
__global__ void gat_gemm_wmma(const float* __restrict__ A,
                              const float* __restrict__ B,
                              float* __restrict__ C,
                              int M) {
    __shared__ float As[16 * LDA];
    const int tid = threadIdx.x;
    const int m0  = blockIdx.x * 16;

    // cooperative load of the 16x128 A tile into LDS
    for (int i = tid; i < 16 * 128; i += (NC / 16) * 32) {
        const int r = i >> 7, c = i & 127;
        const int gr = m0 + r;
        As[r * LDA + c] = (gr < M) ? A[(size_t)gr * 128 + c] : 0.0f;
    }
    __syncthreads();

    const int wave  = tid >> 5;          // one 16x16 N-tile per wave
    const int lane  = tid & 31;
    const int half  = lane >> 4;         // 0: K=0/1 ; 1: K=2/3 (A/B frag layout)
    const int l16   = lane & 15;
    const int koff  = half * 2;
    const int nbase = wave * 16;

    v8f acc = {};
    const float* Arow = As + l16 * LDA + koff;      // A frag row base
    const float* Bp   = B + (size_t)koff * NC + nbase + l16;  // B frag col base
    #pragma unroll
    for (int kb = 0; kb < 32; ++kb) {    // K = 128 in steps of 4
        v2f a, b;
        a.x = Arow[kb * 4];
        a.y = Arow[kb * 4 + 1];
        b.x = Bp[kb * 4 * NC];           // compile-time immediate offsets
        b.y = Bp[kb * 4 * NC + NC];
        acc = __builtin_amdgcn_wmma_f32_16x16x4_f32(
                  false, a, false, b, (short)0, acc, false, false);
    }

    const int col = nbase + l16;
    #pragma unroll
    for (int i = 0; i < 8; ++i) {        // D frag: VGPR i -> row i + 8*half
        const int row = m0 + i + half * 8;
        if (row < M) C[(size_t)row * NC + col] = acc[i];
    }
}

// ---------------- utility fill ----------------
__global__ void gat_fill(float* __restrict__ p, int n, float v) {
    int i = blockIdx.x * blockDim.x + threadIdx.x;
    if (i < n) p[i] = v;
}

// ---------------- per-node attention logits ----------------
template <int H, int CC>
__global__ void gat_alpha(const float* __restrict__ h, const float* __restrict__ asrc,
                          const float* __restrict__ adst, float* __restrict__ al,
                          float* __restrict__ ar, int N) {
    int i = blockIdx.x * blockDim.x + threadIdx.x;
    if (i >= N * H) return;
    const int n = i / H, hh = i - n * H;
    const float* hp = h + (size_t)n * H * CC + hh * CC;
    const float* ap = asrc + hh * CC;
    const float* bp = adst + hh * CC;
    float sl = 0.f, sr = 0.f;
    #pragma unroll 8
    for (int c = 0; c < CC; ++c) { float v = hp[c]; sl += v * ap[c]; sr += v * bp[c]; }
    al[i] = sl; ar[i] = sr;
}

// sign-split float atomic max (works from -inf init)
__device__ inline void atomicMaxFloat(float* addr, float value) {
    if (value >= 0.f) atomicMax((int*)addr, __float_as_int(value));
    else              atomicMin((unsigned int*)addr, __float_as_uint(value));
}

// ---------------- edge pass 1: leaky-relu logits + segment max ----------------
template <int H>
__global__ void gat_edge_max(const int* __restrict__ src, const int* __restrict__ dst,
                             const float* __restrict__ al, const float* __restrict__ ar,
                             float* __restrict__ ebuf, float* __restrict__ m,
                             int E, int N) {
    int e = blockIdx.x * blockDim.x + threadIdx.x;
    if (e >= E + N) return;
    int s, d;
    if (e < E) { s = src[e]; d = dst[e]; } else { s = e - E; d = s; }  // self loops
    #pragma unroll
    for (int h = 0; h < H; ++h) {
        float v = al[s * H + h] + ar[d * H + h];
        v = (v > 0.f) ? v : 0.2f * v;                  // LeakyReLU(0.2)
        ebuf[(size_t)e * H + h] = v;
        atomicMaxFloat(&m[d * H + h], v);
    }
}

// ---------------- edge pass 2: exp + segment sum ----------------
template <int H>
__global__ void gat_edge_exp(const int* __restrict__ dst,
                             float* __restrict__ ebuf, const float* __restrict__ m,
                             float* __restrict__ ssum, int E, int N) {
    int e = blockIdx.x * blockDim.x + threadIdx.x;
    if (e >= E + N) return;
    const int d = (e < E) ? dst[e] : (e - E);
    #pragma unroll
    for (int h = 0; h < H; ++h) {
        float ex = expf(ebuf[(size_t)e * H + h] - m[d * H + h]);
        ebuf[(size_t)e * H + h] = ex;
        atomicAdd(&ssum[d * H + h], ex);
    }
}

// ---------------- edge pass 3: weighted scatter-aggregate (one wave / edge) ----------------
template <int H, int HC>
__global__ void gat_edge_agg(const int* __restrict__ src, const int* __restrict__ dst,
                             const float* __restrict__ ebuf, const float* __restrict__ hfeat,
                             float* __restrict__ agg, int E, int N) {
    int gid  = blockIdx.x * blockDim.x + threadIdx.x;
    int e    = gid >> 5;
    int lane = gid & 31;
    if (e >= E + N) return;
    int s, d;
    if (e < E) { s = src[e]; d = dst[e]; } else { s = e - E; d = s; }
    constexpr int P = HC / 32;            // channels per lane (4 or 2)
    const int c0 = lane * P;
    const int h  = c0 / (HC / H);
    const float w = ebuf[(size_t)e * H + h];
    const float* hp = hfeat + (size_t)s * HC + c0;
    float*       ap = agg   + (size_t)d * HC + c0;
    #pragma unroll
    for (int p = 0; p < P; ++p) atomicAdd(&ap[p], w * hp[p]);
}

// ---------------- softmax normalize + bias (+ optional ReLU), in place ----------------
template <int H, int HC, int DO_RELU>
__global__ void gat_norm(float* __restrict__ agg, const float* __restrict__ ssum,
                         const float* __restrict__ bias, int N) {
    int i = blockIdx.x * blockDim.x + threadIdx.x;
    if (i >= N * HC) return;
    const int n = i / HC, k = i - n * HC;
    const int h = k / (HC / H);
    float v = agg[i] / ssum[n * H + h] + bias[k];
    if (DO_RELU) v = fmaxf(v, 0.f);
    agg[i] = v;
}

// ---------------- decode: sigmoid(dot(z[src], z[dst])), one wave / edge ----------------
__global__ void gat_decode(const int* __restrict__ src, const int* __restrict__ dst,
                           const float* __restrict__ z, float* __restrict__ out, int E) {
    int gid  = blockIdx.x * blockDim.x + threadIdx.x;
    int e    = gid >> 5;
    int lane = gid & 31;
    if (e >= E) return;
    const int s = src[e], d = dst[e];
    const int c = lane * 2;
    float acc = z[(size_t)s * 64 + c]     * z[(size_t)d * 64 + c]
              + z[(size_t)s * 64 + c + 1] * z[(size_t)d * 64 + c + 1];
    for (int off = 16; off > 0; off >>= 1) acc += __shfl_xor(acc, off, 32);
    if (lane == 0) out[e] = 1.f / (1.f + expf(-acc));
}

extern "C" void kernel_launch(void* const* d_in, const int* in_sizes, int n_in,
                              void* d_out, int out_size, void* d_ws, size_t ws_size,
                              hipStream_t stream) {
    const float* x   = (const float*)d_in[0];
    const int*   ei  = (const int*)d_in[1];
    const float* W1  = (const float*)d_in[2];
    const float* a1s = (const float*)d_in[3];
    const float* a1d = (const float*)d_in[4];
    const float* b1  = (const float*)d_in[5];
    const float* W2  = (const float*)d_in[6];
    const float* a2s = (const float*)d_in[7];
    const float* a2d = (const float*)d_in[8];
    const float* b2  = (const float*)d_in[9];
    float* out = (float*)d_out;

    const int N  = in_sizes[0] / 128;
    const int E  = in_sizes[1] / 2;
    const int E2 = E + N;
    const int* src = ei;
    const int* dst = ei + E;

    float* ws = (float*)d_ws;
    size_t off = 0;
    float* h1   = ws + off; off += (size_t)N * 128;
    float* agg1 = ws + off; off += (size_t)N * 128;  // becomes h2 after norm
    float* al1  = ws + off; off += (size_t)N * 4;
    float* ar1  = ws + off; off += (size_t)N * 4;
    float* m1   = ws + off; off += (size_t)N * 4;
    float* s1   = ws + off; off += (size_t)N * 4;
    float* exb  = ws + off; off += (size_t)E2 * 4;   // reused by layer 2
    float* z0   = ws + off; off += (size_t)N * 64;
    float* agg2 = ws + off; off += (size_t)N * 64;   // becomes z after norm
    float* al2  = ws + off; off += (size_t)N;
    float* ar2  = ws + off; off += (size_t)N;
    float* m2   = ws + off; off += (size_t)N;
    float* s2   = ws + off; off += (size_t)N;
    (void)ws_size; (void)n_in; (void)out_size;

    const int TB = 256;
    auto blk = [](long n, int tb) { return (int)((n + tb - 1) / tb); };

    // re-init accumulators every call (harness does not re-poison/zero)
    gat_fill<<<blk((long)N * 128, TB), TB, 0, stream>>>(agg1, N * 128, 0.f);
    gat_fill<<<blk((long)N * 4,   TB), TB, 0, stream>>>(s1,   N * 4,   0.f);
    gat_fill<<<blk((long)N * 4,   TB), TB, 0, stream>>>(m1,   N * 4,  -INFINITY);
    gat_fill<<<blk((long)N * 64,  TB), TB, 0, stream>>>(agg2, N * 64,  0.f);
    gat_fill<<<blk(N, TB), TB, 0, stream>>>(s2, N, 0.f);
    gat_fill<<<blk(N, TB), TB, 0, stream>>>(m2, N, -INFINITY);

    const int mt = (N + 15) / 16;

    // ---- layer 1: GATConv(IN=128 -> 4 heads x 32), ReLU ----
    gat_gemm_wmma<128><<<mt, 256, 0, stream>>>(x, W1, h1, N);
    gat_alpha<4, 32><<<blk((long)N * 4, TB), TB, 0, stream>>>(h1, a1s, a1d, al1, ar1, N);
    gat_edge_max<4><<<blk(E2, TB), TB, 0, stream>>>(src, dst, al1, ar1, exb, m1, E, N);
    gat_edge_exp<4><<<blk(E2, TB), TB, 0, stream>>>(dst, exb, m1, s1, E, N);
    gat_edge_agg<4, 128><<<blk((long)E2 * 32, TB), TB, 0, stream>>>(src, dst, exb, h1, agg1, E, N);
    gat_norm<4, 128, 1><<<blk((long)N * 128, TB), TB, 0, stream>>>(agg1, s1, b1, N);

    // ---- layer 2: GATConv(128 -> 64, 1 head) ----
    gat_gemm_wmma<64><<<mt, 128, 0, stream>>>(agg1, W2, z0, N);
    gat_alpha<1, 64><<<blk(N, TB), TB, 0, stream>>>(z0, a2s, a2d, al2, ar2, N);
    gat_edge_max<1><<<blk(E2, TB), TB, 0, stream>>>(src, dst, al2, ar2, exb, m2, E, N);
    gat_edge_exp<1><<<blk(E2, TB), TB, 0, stream>>>(dst, exb, m2, s2, E, N);
    gat_edge_agg<1, 64><<<blk((long)E2 * 32, TB), TB, 0, stream>>>(src, dst, exb, z0, agg2, E, N);
    gat_norm<1, 64, 0><<<blk((long)N * 64, TB), TB, 0, stream>>>(agg2, s2, b2, N);

    // ---- decode on original edges ----
    gat_decode<<<blk((long)E * 32, TB), TB, 0, stream>>>(src, dst, agg2, out, E);
}